// Attention_5325759447759
// MI455X (gfx1250) — compile-verified
//
#include <hip/hip_runtime.h>

typedef __attribute__((ext_vector_type(2))) float v2f;
typedef __attribute__((ext_vector_type(8))) float v8f;

#define HEADS  8
#define DH     24      // head dim
#define CIN    192
#define C3     576
#define CGI    6       // in channels per group (1x1 qkv / proj)
#define CGO    18      // channels per group (qkv out / dw)
#define KDW    162     // dw-conv GEMM K = 18*9
#define IMG    256
#define TS     8       // spatial tile 8x8
#define TPIX   64
#define HALO   100     // 10x10 halo pixels
#define QKROWS 384     // q+k channels kept in pass 1
#define LP     65      // padded LDS row stride (floats)
#define GRAM   4608    // 8*24*24
#define PSZ    4992    // GRAM + 384 norm partials
#define TPB    4       // tiles per block, pass 1
#define BPB    256     // blocks per image, pass 1 (1024/TPB)

__device__ __forceinline__ v8f wmma4(v2f a, v2f b, v8f c) {
  // D = A(16x4,f32) * B(4x16,f32) + C(16x16,f32)
  return __builtin_amdgcn_wmma_f32_16x16x4_f32(false, a, false, b, (short)0, c,
                                               false, false);
}

__device__ __forceinline__ v8f v8zero() {
  v8f z;
  #pragma unroll
  for (int i = 0; i < 8; ++i) z[i] = 0.f;
  return z;
}

// Implicit-GEMM 3x3 grouped conv on WMMA:
//   D[18,64] = W[18,162] x Im2col[162,64], K padded 162->164.
// wave w: mtile = w>>2, ntile = w&3; 41 K-steps; im2col read straight from t1.
__device__ __forceinline__ v8f dwconv_wmma(const float* __restrict__ wd,
                                           const float* __restrict__ t1,
                                           int wave, int half, int lrow) {
  const int mt = wave >> 2, nt = wave & 3;
  const int m = mt * 16 + lrow;          // out-channel row for A loads
  const int p = nt * 16 + lrow;          // pixel column for B loads
  const int base = (p >> 3) * 10 + (p & 7);
  const bool mv = (m < CGO);
  v8f dacc = v8zero();
  for (int ks = 0; ks < 41; ++ks) {
    const int k0 = ks * 4 + half * 2;
    const int k1 = k0 + 1;
    v2f a, bb;
    a.x = (mv && k0 < KDW) ? wd[m * KDW + k0] : 0.f;
    a.y = (mv && k1 < KDW) ? wd[m * KDW + k1] : 0.f;
    int ci0 = k0 / 9, r0 = k0 - ci0 * 9, dy0 = r0 / 3, dx0 = r0 - dy0 * 3;
    int ci1 = k1 / 9, r1 = k1 - ci1 * 9, dy1 = r1 / 3, dx1 = r1 - dy1 * 3;
    bb.x = (k0 < KDW) ? t1[ci0 * HALO + base + dy0 * 10 + dx0] : 0.f;
    bb.y = (k1 < KDW) ? t1[ci1 * HALO + base + dy1 * 10 + dx1] : 0.f;
    dacc = wmma4(a, bb, dacc);
  }
  return dacc;
}

// ---------------------------------------------------------------------------
// Pass 1: fused grouped 1x1 + 3x3 conv (WMMA implicit GEMM) for q,k channels,
// per-tile raw Gram (WMMA) + per-channel sum-of-squares, accumulated over TPB
// tiles in registers, written as deterministic block partials.
// ---------------------------------------------------------------------------
__global__ __launch_bounds__(256) void k_pass1(const float* __restrict__ x,
                                               const float* __restrict__ wqkv,
                                               const float* __restrict__ wdw,
                                               float* __restrict__ partial) {
  __shared__ float qk[QKROWS * LP];   // q/k channels for the tile
  __shared__ float xs[CGI * HALO];    // x halo slice of one group
  __shared__ float t1[CGO * HALO];    // 1x1-conv result on halo
  __shared__ float wq[CGO * CGI];
  __shared__ float wd[CGO * KDW];

  const int tid  = threadIdx.x;
  const int lane = tid & 31;
  const int wave = tid >> 5;          // == head index in Gram phase
  const int half = lane >> 4;
  const int lrow = lane & 15;
  const int b    = blockIdx.x >> 8;
  const int blk  = blockIdx.x & 255;

  v8f acc[2][2];
  #pragma unroll
  for (int m = 0; m < 2; ++m)
    #pragma unroll
    for (int n = 0; n < 2; ++n) acc[m][n] = v8zero();
  float nacc0 = 0.f, nacc1 = 0.f;

  for (int it = 0; it < TPB; ++it) {
    const int tile = blk * TPB + it;
    const int y0 = (tile >> 5) * TS;
    const int x0 = (tile & 31) * TS;

    for (int g = 0; g < 22; ++g) {   // channels 0..395; keep < 384 (q,k)
      __syncthreads();
      for (int i = tid; i < CGO * CGI; i += 256) wq[i] = wqkv[g * CGO * CGI + i];
      for (int i = tid; i < CGO * KDW; i += 256) wd[i] = wdw[g * CGO * KDW + i];
      if (g + 1 < 22 && tid < CGI)   // global_prefetch_b8 for next group's x
        __builtin_prefetch(&x[(((size_t)b * CIN + (g + 1) * CGI + tid) << 16) +
                              ((size_t)y0 << 8) + x0], 0, 1);
      for (int i = tid; i < CGI * HALO; i += 256) {
        int ci = i / HALO, r = i % HALO, hy = r / 10, hx = r % 10;
        int gy = y0 - 1 + hy, gx = x0 - 1 + hx;
        float v = 0.f;
        if ((unsigned)gy < IMG && (unsigned)gx < IMG)
          v = x[(((size_t)b * CIN + g * CGI + ci) << 16) + ((size_t)gy << 8) + gx];
        xs[i] = v;
      }
      __syncthreads();
      for (int i = tid; i < CGO * HALO; i += 256) {  // 1x1 conv (VALID, no bias)
        int j = i / HALO, r = i % HALO;
        float a = 0.f;
        #pragma unroll
        for (int ci = 0; ci < CGI; ++ci) a += xs[ci * HALO + r] * wq[j * CGI + ci];
        t1[i] = a;
      }
      __syncthreads();
      // 3x3 grouped conv as WMMA implicit GEMM
      v8f dacc = dwconv_wmma(wd, t1, wave, half, lrow);
      const int mt = wave >> 2, nt = wave & 3;
      #pragma unroll
      for (int jj = 0; jj < 8; ++jj) {
        int j = mt * 16 + half * 8 + jj;
        int c = g * CGO + j;
        if (j < CGO && c < QKROWS) qk[c * LP + nt * 16 + lrow] = dacc[jj];
      }
    }
    __syncthreads();

    // per-channel sum of squares (spatial L2 norm, pre-sqrt)
    {
      int c = tid;
      float s = 0.f;
      for (int p = 0; p < TPIX; ++p) { float v = qk[c * LP + p]; s += v * v; }
      nacc0 += s;
      if (tid < QKROWS - 256) {
        c = tid + 256; s = 0.f;
        for (int p = 0; p < TPIX; ++p) { float v = qk[c * LP + p]; s += v * v; }
        nacc1 += s;
      }
    }

    // Gram: G[d,e] += sum_s q[h,d,s]*k[h,e,s]; one head per wave, K=64 pixels
    for (int ks = 0; ks < 16; ++ks) {
      const int kb = ks * 4 + half * 2;
      v2f a0, a1, b0, b1;
      int r0 = wave * DH + lrow;                    // A rows, m-tile 0
      int r1 = wave * DH + 16 + lrow;               // m-tile 1 (pad rows ignored)
      a0.x = qk[r0 * LP + kb];  a0.y = qk[r0 * LP + kb + 1];
      a1.x = qk[r1 * LP + kb];  a1.y = qk[r1 * LP + kb + 1];
      int c0 = 192 + wave * DH + lrow;              // B = k^T, n-tile 0
      int c1 = 192 + wave * DH + 16 + lrow;         // n-tile 1
      if (c1 > QKROWS - 1) c1 = QKROWS - 1;         // pad cols ignored on store
      b0.x = qk[c0 * LP + kb];  b0.y = qk[c0 * LP + kb + 1];
      b1.x = qk[c1 * LP + kb];  b1.y = qk[c1 * LP + kb + 1];
      acc[0][0] = wmma4(a0, b0, acc[0][0]);
      acc[0][1] = wmma4(a0, b1, acc[0][1]);
      acc[1][0] = wmma4(a1, b0, acc[1][0]);
      acc[1][1] = wmma4(a1, b1, acc[1][1]);
    }
    __syncthreads();   // before next tile overwrites qk
  }

  const size_t pb = ((size_t)b * BPB + blk) * PSZ;
  #pragma unroll
  for (int mt = 0; mt < 2; ++mt)
    #pragma unroll
    for (int nt = 0; nt < 2; ++nt)
      #pragma unroll
      for (int j = 0; j < 8; ++j) {
        int d = mt * 16 + half * 8 + j;
        int e = nt * 16 + lrow;
        if (d < DH && e < DH)
          partial[pb + wave * (DH * DH) + d * DH + e] = acc[mt][nt][j];
      }
  partial[pb + GRAM + tid] = nacc0;
  if (tid < QKROWS - 256) partial[pb + GRAM + 256 + tid] = nacc1;
}

// ---------------------------------------------------------------------------
// Deterministic reduction of block partials.
// ---------------------------------------------------------------------------
__global__ void k_reduce(const float* __restrict__ partial,
                         float* __restrict__ reduced) {
  int i = blockIdx.x * blockDim.x + threadIdx.x;
  if (i >= 2 * PSZ) return;
  int b = i / PSZ, j = i % PSZ;
  float s = 0.f;
  for (int blk = 0; blk < BPB; ++blk) s += partial[((size_t)b * BPB + blk) * PSZ + j];
  reduced[i] = s;
}

// ---------------------------------------------------------------------------
// attn = softmax(G / (||q_d|| * ||k_e||) * temperature)
// ---------------------------------------------------------------------------
__global__ __launch_bounds__(32) void k_attn(const float* __restrict__ reduced,
                                             const float* __restrict__ temp,
                                             float* __restrict__ attn) {
  const int b = blockIdx.x >> 3, h = blockIdx.x & 7;
  const int d = threadIdx.x;
  __shared__ float nk[DH];
  __shared__ float nq[DH];
  if (d < DH) {
    nq[d] = fmaxf(sqrtf(reduced[b * PSZ + GRAM + h * DH + d]), 1e-12f);
    nk[d] = fmaxf(sqrtf(reduced[b * PSZ + GRAM + 192 + h * DH + d]), 1e-12f);
  }
  __syncthreads();
  if (d < DH) {
    const float t = temp[h];
    float row[DH], mx = -3.4e38f;
    for (int e = 0; e < DH; ++e) {
      float g = reduced[b * PSZ + h * (DH * DH) + d * DH + e];
      float a = g / (nq[d] * nk[e]) * t;
      row[e] = a; mx = fmaxf(mx, a);
    }
    float s = 0.f;
    for (int e = 0; e < DH; ++e) { row[e] = expf(row[e] - mx); s += row[e]; }
    float inv = 1.f / s;
    for (int e = 0; e < DH; ++e)
      attn[((b * HEADS + h) * DH + d) * DH + e] = row[e] * inv;
  }
}

// ---------------------------------------------------------------------------
// Pass 2: recompute v channels (groups 21..31, WMMA implicit-GEMM conv),
// out = attn @ v (WMMA), fused grouped 1x1 output projection.
// ---------------------------------------------------------------------------
__global__ __launch_bounds__(256) void k_pass2(const float* __restrict__ x,
                                               const float* __restrict__ wqkv,
                                               const float* __restrict__ wdw,
                                               const float* __restrict__ wproj,
                                               const float* __restrict__ attn,
                                               float* __restrict__ out) {
  __shared__ float vbuf[CIN * LP];
  __shared__ float obuf[CIN * LP];
  __shared__ float at[GRAM];
  __shared__ float xs[CGI * HALO];
  __shared__ float t1[CGO * HALO];
  __shared__ float wq[CGO * CGI];
  __shared__ float wd[CGO * KDW];
  __shared__ float wp[CIN * CGI];

  const int tid  = threadIdx.x;
  const int lane = tid & 31;
  const int wave = tid >> 5;          // head in attn@v phase
  const int half = lane >> 4;
  const int lrow = lane & 15;
  const int b    = blockIdx.x >> 10;
  const int tile = blockIdx.x & 1023;
  const int y0 = (tile >> 5) * TS;
  const int x0 = (tile & 31) * TS;

  for (int i = tid; i < GRAM; i += 256) at[i] = attn[b * GRAM + i];
  for (int i = tid; i < CIN * CGI; i += 256) wp[i] = wproj[i];

  for (int g = 21; g < 32; ++g) {     // channels 378..575; keep >= 384 (v)
    __syncthreads();
    for (int i = tid; i < CGO * CGI; i += 256) wq[i] = wqkv[g * CGO * CGI + i];
    for (int i = tid; i < CGO * KDW; i += 256) wd[i] = wdw[g * CGO * KDW + i];
    for (int i = tid; i < CGI * HALO; i += 256) {
      int ci = i / HALO, r = i % HALO, hy = r / 10, hx = r % 10;
      int gy = y0 - 1 + hy, gx = x0 - 1 + hx;
      float v = 0.f;
      if ((unsigned)gy < IMG && (unsigned)gx < IMG)
        v = x[(((size_t)b * CIN + g * CGI + ci) << 16) + ((size_t)gy << 8) + gx];
      xs[i] = v;
    }
    __syncthreads();
    for (int i = tid; i < CGO * HALO; i += 256) {
      int j = i / HALO, r = i % HALO;
      float a = 0.f;
      #pragma unroll
      for (int ci = 0; ci < CGI; ++ci) a += xs[ci * HALO + r] * wq[j * CGI + ci];
      t1[i] = a;
    }
    __syncthreads();
    v8f dacc = dwconv_wmma(wd, t1, wave, half, lrow);
    const int mt = wave >> 2, nt = wave & 3;
    #pragma unroll
    for (int jj = 0; jj < 8; ++jj) {
      int j = mt * 16 + half * 8 + jj;
      int c = g * CGO + j;
      if (j < CGO && c >= QKROWS) vbuf[(c - QKROWS) * LP + nt * 16 + lrow] = dacc[jj];
    }
  }
  __syncthreads();

  // out[h,d,s] = sum_e attn[h,d,e] * v[h,e,s]; M=24(pad32), K=24, N=64
  v8f acc[2][4];
  #pragma unroll
  for (int m = 0; m < 2; ++m)
    #pragma unroll
    for (int n = 0; n < 4; ++n) acc[m][n] = v8zero();

  for (int ks = 0; ks < 6; ++ks) {
    const int kb = ks * 4 + half * 2;   // e in 0..23
    v2f a0, a1;
    int d0 = lrow;
    int d1 = 16 + lrow; if (d1 > DH - 1) d1 = DH - 1;   // pad rows ignored
    a0.x = at[wave * (DH * DH) + d0 * DH + kb];
    a0.y = at[wave * (DH * DH) + d0 * DH + kb + 1];
    a1.x = at[wave * (DH * DH) + d1 * DH + kb];
    a1.y = at[wave * (DH * DH) + d1 * DH + kb + 1];
    #pragma unroll
    for (int nt = 0; nt < 4; ++nt) {
      v2f bb;
      int s = nt * 16 + lrow;
      bb.x = vbuf[(wave * DH + kb) * LP + s];
      bb.y = vbuf[(wave * DH + kb + 1) * LP + s];
      acc[0][nt] = wmma4(a0, bb, acc[0][nt]);
      acc[1][nt] = wmma4(a1, bb, acc[1][nt]);
    }
  }
  #pragma unroll
  for (int mt = 0; mt < 2; ++mt)
    #pragma unroll
    for (int nt = 0; nt < 4; ++nt)
      #pragma unroll
      for (int j = 0; j < 8; ++j) {
        int d = mt * 16 + half * 8 + j;
        if (d < DH) {
          int s = nt * 16 + lrow;
          obuf[(wave * DH + d) * LP + s] = acc[mt][nt][j];
        }
      }
  __syncthreads();

  // grouped 1x1 output projection (6 in -> 6 out per group) + store
  for (int i = tid; i < CIN * TPIX; i += 256) {
    int co = i / TPIX, p = i % TPIX;
    int grp = co / CGI;
    float a = 0.f;
    #pragma unroll
    for (int ci = 0; ci < CGI; ++ci)
      a += obuf[(grp * CGI + ci) * LP + p] * wp[co * CGI + ci];
    int py = p >> 3, px = p & 7;
    out[(((size_t)b * CIN + co) << 16) + ((size_t)(y0 + py) << 8) + (x0 + px)] = a;
  }
}

// ---------------------------------------------------------------------------
extern "C" void kernel_launch(void* const* d_in, const int* in_sizes, int n_in,
                              void* d_out, int out_size, void* d_ws, size_t ws_size,
                              hipStream_t stream) {
  const float* x     = (const float*)d_in[0];
  const float* wqkv  = (const float*)d_in[1];
  const float* wdw   = (const float*)d_in[2];
  const float* wproj = (const float*)d_in[3];
  const float* temp  = (const float*)d_in[4];
  float* out = (float*)d_out;

  float* ws      = (float*)d_ws;
  float* partial = ws;                               // 2*BPB*PSZ floats (~10.2 MB)
  float* reduced = partial + (size_t)2 * BPB * PSZ;  // 2*PSZ
  float* attnw   = reduced + (size_t)2 * PSZ;        // 2*GRAM

  k_pass1<<<dim3(2 * BPB), dim3(256), 0, stream>>>(x, wqkv, wdw, partial);
  k_reduce<<<dim3((2 * PSZ + 255) / 256), dim3(256), 0, stream>>>(partial, reduced);
  k_attn<<<dim3(2 * HEADS), dim3(32), 0, stream>>>(reduced, temp, attnw);
  k_pass2<<<dim3(2 * 1024), dim3(256), 0, stream>>>(x, wqkv, wdw, wproj, attnw, out);
}